// GNN_52415780880534
// MI455X (gfx1250) — compile-verified
//
#include <hip/hip_runtime.h>
#include <stdint.h>

#define NN   50000   // nodes
#define NE   800000  // edges
#define CH   128     // in/hidden dim
#define NG   64      // graphs
#define NOUT 16      // output dim

#define LDSW 132     // padded LDS row stride in floats (128 + 4) -> conflict-free b64 reads

typedef __attribute__((ext_vector_type(2))) float v2f;
typedef __attribute__((ext_vector_type(8))) float v8f;

// ---------------- degree / normalization ----------------

__global__ __launch_bounds__(256) void k_deg_init(float* __restrict__ deg) {
    int i = blockIdx.x * 256 + threadIdx.x;
    if (i < NN) deg[i] = 1.0f;  // self-loop contributes 1 to every node's degree
}

__global__ __launch_bounds__(256) void k_deg_scatter(const long long* __restrict__ dst,
                                                     float* __restrict__ deg) {
    int e = blockIdx.x * 256 + threadIdx.x;
    if (e < NE) atomicAdd(&deg[(int)dst[e]], 1.0f);
}

__global__ __launch_bounds__(256) void k_rsqrt_inplace(float* __restrict__ deg) {
    int i = blockIdx.x * 256 + threadIdx.x;
    if (i < NN) deg[i] = rsqrtf(deg[i]);  // deg >= 1 always (self-loops)
}

// ---------------- dense GEMM via fp32 WMMA + async LDS staging ----------------
// H[NN x 128] = X[NN x 128] * W[128 x 128].
// One 256-thread block (8 waves) per 16-row tile of X:
//   - stage the 16x128 fp32 X-slice into LDS once via global_load_async_to_lds_b128
//   - wave w computes column tile w (16 cols) with V_WMMA_F32_16X16X4_F32, K stepped by 4
__global__ __launch_bounds__(256) void k_gemm_f32_wmma(const float* __restrict__ X,
                                                       const float* __restrict__ W,
                                                       float* __restrict__ H) {
    __shared__ float xs[16 * LDSW];

    const int tm   = blockIdx.x;         // row tile index (0 .. NN/16-1)
    const int wv   = threadIdx.x >> 5;   // column tile (0..7)
    const int lane = threadIdx.x & 31;

    // --- async global -> LDS staging of the X tile (8192 B, 32 B per thread) ---
    {
        const int r  = threadIdx.x >> 4;        // row within tile (0..15)
        const int c8 = (threadIdx.x & 15) * 8;  // starting float column (0,8,...,120)
        uint64_t gaddr = (uint64_t)(uintptr_t)(X + (size_t)(tm * 16 + r) * CH + c8);
        uint32_t laddr = (uint32_t)(uintptr_t)&xs[r * LDSW + c8];
        // INST_OFFSET applies to both global and LDS addresses (ISA 15.18.3):
        asm volatile("global_load_async_to_lds_b128 %0, %1, off"
                     :: "v"(laddr), "v"(gaddr) : "memory");
        asm volatile("global_load_async_to_lds_b128 %0, %1, off offset:16"
                     :: "v"(laddr), "v"(gaddr) : "memory");
        asm volatile("s_wait_asynccnt 0x0" ::: "memory");
    }
    __syncthreads();

    // --- WMMA compute ---
    const int half = lane >> 4;   // 0: K-pair {0,1}, 1: K-pair {2,3}
    const int sub  = lane & 15;   // M (for A) / N (for B,C,D)
    const int koff = half * 2;
    const int col  = wv * 16 + sub;

    const float* arow = &xs[sub * LDSW];  // LDS row for this lane's M

    v8f acc = {};
#pragma unroll 4
    for (int k = 0; k < CH; k += 4) {
        // A 16x4 fp32 fragment from LDS (ds_load_b64, conflict-free via LDSW pad)
        float2 av = *reinterpret_cast<const float2*>(arow + k + koff);
        v2f a; a.x = av.x; a.y = av.y;
        // B 4x16 fp32 fragment from global (L2-resident 64 KB W)
        v2f b;
        b.x = W[(size_t)(k + koff)     * CH + col];
        b.y = W[(size_t)(k + koff + 1) * CH + col];
        acc = __builtin_amdgcn_wmma_f32_16x16x4_f32(
            /*neg_a=*/false, a, /*neg_b=*/false, b,
            /*c_mod=*/(short)0, acc, /*reuse_a=*/false, /*reuse_b=*/false);
    }
    // D 16x16 fp32: VGPR r -> row (half*8 + r), col = sub
    float* hp = H + (size_t)(tm * 16 + half * 8) * CH + col;
#pragma unroll
    for (int r = 0; r < 8; ++r)
        hp[(size_t)r * CH] = acc[r];
}

// ---------------- aggregation ----------------

// agg[i][c] = h[i][c] * dis[i]^2  (the self-loop edge, non-atomic seed)
__global__ __launch_bounds__(256) void k_selfloop_seed(const float* __restrict__ h,
                                                       const float* __restrict__ dis,
                                                       float* __restrict__ agg) {
    long long idx = (long long)blockIdx.x * 256 + threadIdx.x;
    if (idx >= (long long)NN * CH) return;
    int node = (int)(idx >> 7);  // CH == 128
    float w = dis[node] * dis[node];
    agg[idx] = h[idx] * w;
}

// one wave per edge: lane handles 4 channels (float4 gather + 4 float atomics)
__global__ __launch_bounds__(256) void k_edge_scatter(const long long* __restrict__ src,
                                                      const long long* __restrict__ dst,
                                                      const float* __restrict__ dis,
                                                      const float* __restrict__ h,
                                                      float* __restrict__ agg) {
    int wave = (blockIdx.x * 256 + threadIdx.x) >> 5;
    int lane = threadIdx.x & 31;
    if (wave >= NE) return;
    int s = (int)src[wave];
    int d = (int)dst[wave];
    float nrm = dis[s] * dis[d];
    float4 v = reinterpret_cast<const float4*>(h + (size_t)s * CH)[lane];
    float* ad = agg + (size_t)d * CH + lane * 4;
    atomicAdd(ad + 0, v.x * nrm);
    atomicAdd(ad + 1, v.y * nrm);
    atomicAdd(ad + 2, v.z * nrm);
    atomicAdd(ad + 3, v.w * nrm);
}

__global__ __launch_bounds__(256) void k_bias_relu(const float* __restrict__ agg,
                                                   const float* __restrict__ bias,
                                                   float* __restrict__ out) {
    long long idx = (long long)blockIdx.x * 256 + threadIdx.x;
    if (idx >= (long long)NN * CH) return;
    float v = agg[idx] + bias[(int)(idx & (CH - 1))];
    out[idx] = v > 0.0f ? v : 0.0f;
}

// ---------------- pooling + FC ----------------

__global__ __launch_bounds__(256) void k_pool_zero(float* __restrict__ pool,
                                                   float* __restrict__ cnt) {
    int i = blockIdx.x * 256 + threadIdx.x;
    if (i < NG * CH) pool[i] = 0.0f;
    if (i < NG) cnt[i] = 0.0f;
}

// one wave per node
__global__ __launch_bounds__(256) void k_pool(const float* __restrict__ feat,
                                              const long long* __restrict__ batch,
                                              float* __restrict__ pool,
                                              float* __restrict__ cnt) {
    int wave = (blockIdx.x * 256 + threadIdx.x) >> 5;
    int lane = threadIdx.x & 31;
    if (wave >= NN) return;
    int g = (int)batch[wave];
    float4 v = reinterpret_cast<const float4*>(feat + (size_t)wave * CH)[lane];
    float* pg = pool + (size_t)g * CH + lane * 4;
    atomicAdd(pg + 0, v.x);
    atomicAdd(pg + 1, v.y);
    atomicAdd(pg + 2, v.z);
    atomicAdd(pg + 3, v.w);
    if (lane == 0) atomicAdd(&cnt[g], 1.0f);
}

// out[g][o] = (sum_c pool[g][c] * Wfc[c][o]) / max(cnt[g],1) + bfc[o]
__global__ __launch_bounds__(256) void k_fc(const float* __restrict__ pool,
                                            const float* __restrict__ cnt,
                                            const float* __restrict__ Wfc,
                                            const float* __restrict__ bfc,
                                            float* __restrict__ out) {
    int idx = blockIdx.x * 256 + threadIdx.x;
    if (idx >= NG * NOUT) return;
    int g = idx >> 4;
    int o = idx & 15;
    float acc = 0.0f;
#pragma unroll 4
    for (int c = 0; c < CH; ++c)
        acc += pool[g * CH + c] * Wfc[c * NOUT + o];
    float inv = 1.0f / fmaxf(cnt[g], 1.0f);
    out[idx] = acc * inv + bfc[o];
}

// ---------------- launcher ----------------

extern "C" void kernel_launch(void* const* d_in, const int* in_sizes, int n_in,
                              void* d_out, int out_size, void* d_ws, size_t ws_size,
                              hipStream_t stream) {
    const float*     x     = (const float*)d_in[0];
    const long long* ei    = (const long long*)d_in[1];  // [2, NE] int64
    const long long* batch = (const long long*)d_in[2];  // [NN] int64
    const float*     W1    = (const float*)d_in[3];
    const float*     b1    = (const float*)d_in[4];
    const float*     W2    = (const float*)d_in[5];
    const float*     b2    = (const float*)d_in[6];
    const float*     Wfc   = (const float*)d_in[7];
    const float*     bfc   = (const float*)d_in[8];
    float*           out   = (float*)d_out;

    const long long* src = ei;       // row 0
    const long long* dst = ei + NE;  // row 1

    // workspace layout (floats): deg/dis [NN] | h [NN*CH] | agg [NN*CH] | feat [NN*CH] | pool [NG*CH] | cnt [NG]
    float* ws   = (float*)d_ws;
    float* deg  = ws;
    float* hbuf = deg + NN;
    float* agg  = hbuf + (size_t)NN * CH;
    float* feat = agg + (size_t)NN * CH;
    float* pool = feat + (size_t)NN * CH;
    float* cnt  = pool + (size_t)NG * CH;

    const int nodeBlocks     = (NN + 255) / 256;
    const int edgeBlocks     = (NE + 255) / 256;
    const int ncBlocks       = (int)(((long long)NN * CH + 255) / 256);
    const int gemmBlocks     = NN / 16;      // one block per 16-row tile
    const int edgeWaveBlocks = (NE + 7) / 8; // 8 waves per 256-thread block
    const int poolWaveBlocks = (NN + 7) / 8;

    // normalization
    k_deg_init<<<nodeBlocks, 256, 0, stream>>>(deg);
    k_deg_scatter<<<edgeBlocks, 256, 0, stream>>>(dst, deg);
    k_rsqrt_inplace<<<nodeBlocks, 256, 0, stream>>>(deg);  // deg now holds deg_inv_sqrt

    // layer 1
    k_gemm_f32_wmma<<<gemmBlocks, 256, 0, stream>>>(x, W1, hbuf);
    k_selfloop_seed<<<ncBlocks, 256, 0, stream>>>(hbuf, deg, agg);
    k_edge_scatter<<<edgeWaveBlocks, 256, 0, stream>>>(src, dst, deg, hbuf, agg);
    k_bias_relu<<<ncBlocks, 256, 0, stream>>>(agg, b1, feat);

    // layer 2
    k_gemm_f32_wmma<<<gemmBlocks, 256, 0, stream>>>(feat, W2, hbuf);
    k_selfloop_seed<<<ncBlocks, 256, 0, stream>>>(hbuf, deg, agg);
    k_edge_scatter<<<edgeWaveBlocks, 256, 0, stream>>>(src, dst, deg, hbuf, agg);
    k_bias_relu<<<ncBlocks, 256, 0, stream>>>(agg, b2, feat);

    // mean pool + FC
    k_pool_zero<<<(NG * CH + 255) / 256, 256, 0, stream>>>(pool, cnt);
    k_pool<<<poolWaveBlocks, 256, 0, stream>>>(feat, batch, pool, cnt);
    k_fc<<<(NG * NOUT + 255) / 256, 256, 0, stream>>>(pool, cnt, Wfc, bfc, out);
}